// PersistentGraphAlignmentLoss_56573309223818
// MI455X (gfx1250) — compile-verified
//
#include <hip/hip_runtime.h>
#include <hip/hip_bf16.h>
#include <float.h>

// ---------------------------------------------------------------------------
// PersistentGraphAlignmentLoss for gfx1250 (MI455X)
//   D = pairwise dist via V_WMMA_F32_16X16X4_F32 Gram matrix
//   Prim MST (single-WGP LDS-resident), H1 persistence,
//   bitonic sort with LDS-fused low-j passes + async global<->LDS movement,
//   deterministic 1-Wasserstein reduction.
// ---------------------------------------------------------------------------

typedef __attribute__((ext_vector_type(2))) float v2f;
typedef __attribute__((ext_vector_type(8))) float v8f;
typedef __attribute__((address_space(3))) float as3_float;

#define N_PTS    4096
#define D_FEAT   128
#define NPAIR    8386560u      // 4096*4095/2
#define NPAD     8388608u      // 2^23
#define TREE_U32 524288u       // 4096*4096/32
#define CHUNK    2048u         // elements per LDS-resident sort chunk (8 KB)

#define USE_ASYNC_LDS 1        // global_{load,store}_async_*_lds via inline asm

// ---------------- row squared norms ----------------
__global__ __launch_bounds__(256)
void row_sq_kernel(const float* __restrict__ X, float* __restrict__ sq)
{
    int i = blockIdx.x * 256 + threadIdx.x;
    if (i < N_PTS) {
        const float* r = X + (size_t)i * D_FEAT;
        float s = 0.f;
        #pragma unroll 8
        for (int k = 0; k < D_FEAT; ++k) { float v = r[k]; s += v * v; }
        sq[i] = s;
    }
}

// ---------------- Gram + distance via WMMA f32 16x16x4 ----------------
// One wave per 16x16 output tile; 8 waves (256 threads) per block.
// A layout (16x4 f32): lanes 0-15 -> M=lane, K={0,1}; lanes 16-31 -> K={2,3}.
// C/D layout (16x16 f32): vgpr v, lanes 0-15 -> M=v, N=lane;
//                         lanes 16-31 -> M=v+8, N=lane-16.
__global__ __launch_bounds__(256)
void gram_dist_kernel(const float* __restrict__ X,
                      const float* __restrict__ sq,
                      float* __restrict__ D,
                      unsigned* __restrict__ capu)
{
    const int wave   = threadIdx.x >> 5;
    const int lane   = threadIdx.x & 31;
    const int tileId = blockIdx.x * 8 + wave;
    const int tpr    = N_PTS >> 4;            // 256 tiles per row
    const int tr     = tileId / tpr;
    const int tc     = tileId % tpr;
    const int r0     = tr << 4;
    const int c0     = tc << 4;
    const int m      = lane & 15;
    const int sel    = lane >> 4;             // 0: K=0,1   1: K=2,3

    const float* Ar = X + (size_t)(r0 + m) * D_FEAT + sel * 2;
    const float* Br = X + (size_t)(c0 + m) * D_FEAT + sel * 2;

    v8f acc = {};
    #pragma unroll 4
    for (int kc = 0; kc < D_FEAT; kc += 4) {
        v2f a, b;
        a.x = Ar[kc];     a.y = Ar[kc + 1];
        b.x = Br[kc];     b.y = Br[kc + 1];
        acc = __builtin_amdgcn_wmma_f32_16x16x4_f32(
            /*neg_a=*/false, a, /*neg_b=*/false, b,
            /*c_mod=*/(short)0, acc, /*reuse_a=*/false, /*reuse_b=*/false);
    }

    const int col = c0 + m;
    const float sqc = sq[col];
    float lmax = 0.f;
    #pragma unroll
    for (int v = 0; v < 8; ++v) {
        int row = r0 + v + sel * 8;
        float d2 = sq[row] + sqc - 2.0f * acc[v];
        d2 = fmaxf(d2, 0.f);
        float d = (d2 > 1e-12f) ? sqrtf(d2) : 0.f;
        D[(size_t)row * N_PTS + col] = d;
        lmax = fmaxf(lmax, d);
    }

    __shared__ float smax[256];
    smax[threadIdx.x] = lmax;
    __syncthreads();
    for (int s = 128; s > 0; s >>= 1) {
        if (threadIdx.x < s)
            smax[threadIdx.x] = fmaxf(smax[threadIdx.x], smax[threadIdx.x + s]);
        __syncthreads();
    }
    if (threadIdx.x == 0)
        atomicMax(capu, __float_as_uint(smax[0]));   // all d >= 0: uint order == float order
}

// ---------------- Prim MST (single block, LDS-resident state) ----------------
__global__ __launch_bounds__(1024)
void prim_kernel(const float* __restrict__ D, unsigned* __restrict__ tree)
{
    __shared__ float mind[N_PTS];
    __shared__ int   par[N_PTS];
    __shared__ float rv[1024];
    __shared__ int   ri[1024];

    const int tid = threadIdx.x;
    for (int i = tid; i < N_PTS; i += 1024) { mind[i] = D[i]; par[i] = 0; }
    __syncthreads();
    if (tid == 0) mind[0] = -1.0f;    // visited marker (distances >= 0)
    __syncthreads();

    for (int step = 0; step < N_PTS - 1; ++step) {
        // argmin with first-index tie-break (matches jnp.argmin)
        float bv = FLT_MAX; int bi = N_PTS;
        for (int i = tid; i < N_PTS; i += 1024) {
            float v = mind[i];
            float vv = (v < 0.f) ? FLT_MAX : v;
            if (vv < bv || (vv == bv && i < bi)) { bv = vv; bi = i; }
        }
        rv[tid] = bv; ri[tid] = bi;
        __syncthreads();
        for (int s = 512; s > 0; s >>= 1) {
            if (tid < s) {
                float ov = rv[tid + s]; int oi = ri[tid + s];
                if (ov < rv[tid] || (ov == rv[tid] && oi < ri[tid])) {
                    rv[tid] = ov; ri[tid] = oi;
                }
            }
            __syncthreads();
        }
        const int j = ri[0];
        if (tid == 0) {
            int pj = par[j];
            int r = pj < j ? pj : j;
            int c = pj < j ? j  : pj;
            size_t e = (size_t)r * N_PTS + c;
            atomicOr(&tree[e >> 5], 1u << (unsigned)(e & 31));
            mind[j] = -1.0f;
        }
        __syncthreads();
        const float* row = D + (size_t)j * N_PTS;
        for (int i = tid; i < N_PTS; i += 1024) {
            float mcur = mind[i];
            if (mcur >= 0.f) {
                float dji = row[i];
                if (dji < mcur) { mind[i] = dji; par[i] = j; }
            }
        }
        __syncthreads();
    }
}

// ---------------- persistence values (upper triangle) ----------------
__global__ __launch_bounds__(256)
void persistence_kernel(const float* __restrict__ D,
                        const unsigned* __restrict__ tree,
                        const unsigned* __restrict__ capu,
                        float* __restrict__ p)
{
    unsigned idx = blockIdx.x * 256u + threadIdx.x;   // over full n*n
    unsigned i = idx >> 12;                           // / 4096
    unsigned j = idx & 4095u;
    if (j > i) {
        unsigned long long t =
            (unsigned long long)i * (unsigned long long)(2u * N_PTS - i - 1u) / 2ull
            + (j - i - 1u);
        size_t e = (size_t)i * N_PTS + j;
        bool is_tree = (tree[e >> 5] >> (unsigned)(e & 31)) & 1u;
        float cap = __uint_as_float(*capu);
        p[t] = is_tree ? 0.f : (cap - D[e]);
    }
}

// ---------------- helpers: zero fills ----------------
__global__ __launch_bounds__(256)
void zero_u32_kernel(unsigned* __restrict__ p, unsigned count)
{
    unsigned i = blockIdx.x * 256u + threadIdx.x;
    if (i < count) p[i] = 0u;
}

__global__ __launch_bounds__(256)
void pad_zero_kernel(float* __restrict__ p, unsigned start, unsigned end)
{
    unsigned i = start + blockIdx.x * 256u + threadIdx.x;
    if (i < end) p[i] = 0.f;   // pads sort to front of both arrays -> cancel
}

// ---------------- bitonic sort: global pass (j >= CHUNK) ----------------
__global__ __launch_bounds__(256)
void bitonic_kernel(float* __restrict__ a, unsigned k, unsigned j)
{
    unsigned i = blockIdx.x * 256u + threadIdx.x;
    unsigned l = i ^ j;
    if (l > i) {
        float ai = a[i], al = a[l];
        bool up = ((i & k) == 0u);
        if ((ai > al) == up) { a[i] = al; a[l] = ai; }
    }
}

// ---- LDS chunk movement: async global<->LDS (ASYNCcnt) with fallback ----
__device__ __forceinline__ void load_chunk(const float* __restrict__ a,
                                           unsigned base, float* tile)
{
#if USE_ASYNC_LDS
    #pragma unroll
    for (unsigned t = threadIdx.x * 4u; t < CHUNK; t += 1024u) {
        unsigned laddr = (unsigned)(uintptr_t)(as3_float*)&tile[t];
        const float* gp = a + base + t;
        asm volatile("global_load_async_to_lds_b128 %0, %1, off"
                     :: "v"(laddr), "v"(gp) : "memory");
    }
    asm volatile("s_wait_asynccnt 0x0" ::: "memory");
#else
    for (unsigned t = threadIdx.x; t < CHUNK; t += 256u) tile[t] = a[base + t];
#endif
    __syncthreads();
}

__device__ __forceinline__ void store_chunk(float* __restrict__ a,
                                            unsigned base, float* tile)
{
    __syncthreads();
#if USE_ASYNC_LDS
    #pragma unroll
    for (unsigned t = threadIdx.x * 4u; t < CHUNK; t += 1024u) {
        unsigned laddr = (unsigned)(uintptr_t)(as3_float*)&tile[t];
        float* gp = a + base + t;
        asm volatile("global_store_async_from_lds_b128 %0, %1, off"
                     :: "v"(gp), "v"(laddr) : "memory");
    }
    asm volatile("s_wait_asynccnt 0x0" ::: "memory");
#else
    for (unsigned t = threadIdx.x; t < CHUNK; t += 256u) a[base + t] = tile[t];
#endif
}

__device__ __forceinline__ void lds_pass(float* tile, unsigned base,
                                         unsigned k, unsigned j)
{
    for (unsigned t = threadIdx.x; t < CHUNK; t += 256u) {
        unsigned l = t ^ j;                      // j < CHUNK: partner in-chunk
        if (l > t) {
            float x0 = tile[t], x1 = tile[l];
            bool up = (((base + t) & k) == 0u);  // direction from GLOBAL index
            if ((x0 > x1) == up) { tile[t] = x1; tile[l] = x0; }
        }
    }
    __syncthreads();
}

// Fuses ALL passes k=2..CHUNK (j<k) — identical network to the reference.
__global__ __launch_bounds__(256)
void bitonic_local_sort_kernel(float* __restrict__ a)
{
    __shared__ float tile[CHUNK];
    const unsigned base = blockIdx.x * CHUNK;
    load_chunk(a, base, tile);
    for (unsigned k = 2; k <= CHUNK; k <<= 1)
        for (unsigned j = k >> 1; j > 0; j >>= 1)
            lds_pass(tile, base, k, j);
    store_chunk(a, base, tile);
}

// Fuses passes j=CHUNK/2..1 for a fixed k (k > CHUNK).
__global__ __launch_bounds__(256)
void bitonic_local_merge_kernel(float* __restrict__ a, unsigned k)
{
    __shared__ float tile[CHUNK];
    const unsigned base = blockIdx.x * CHUNK;
    load_chunk(a, base, tile);
    for (unsigned j = CHUNK >> 1; j > 0; j >>= 1)
        lds_pass(tile, base, k, j);
    store_chunk(a, base, tile);
}

// ---------------- deterministic Wasserstein reduction ----------------
__global__ __launch_bounds__(256)
void wass_partial_kernel(const float* __restrict__ p1,
                         const float* __restrict__ p2,
                         float* __restrict__ partials)
{
    __shared__ float s[256];
    const float4* q1 = (const float4*)p1;
    const float4* q2 = (const float4*)p2;
    float acc = 0.f;
    for (unsigned i = blockIdx.x * 256u + threadIdx.x; i < NPAD / 4u;
         i += gridDim.x * 256u) {
        float4 a = q1[i], b = q2[i];
        acc += fabsf(a.x - b.x) + fabsf(a.y - b.y) +
               fabsf(a.z - b.z) + fabsf(a.w - b.w);
    }
    s[threadIdx.x] = acc;
    __syncthreads();
    for (int st = 128; st > 0; st >>= 1) {
        if (threadIdx.x < st) s[threadIdx.x] += s[threadIdx.x + st];
        __syncthreads();
    }
    if (threadIdx.x == 0) partials[blockIdx.x] = s[0];
}

__global__ __launch_bounds__(1024)
void wass_final_kernel(const float* __restrict__ partials,
                       float* __restrict__ out, int nblocks, float weight)
{
    __shared__ float s[1024];
    float acc = 0.f;
    for (int i = threadIdx.x; i < nblocks; i += 1024) acc += partials[i];
    s[threadIdx.x] = acc;
    __syncthreads();
    for (int st = 512; st > 0; st >>= 1) {
        if (threadIdx.x < st) s[threadIdx.x] += s[threadIdx.x + st];
        __syncthreads();
    }
    if (threadIdx.x == 0) out[0] = weight * s[0];
}

// ---------------------------------------------------------------------------
extern "C" void kernel_launch(void* const* d_in, const int* in_sizes, int n_in,
                              void* d_out, int out_size, void* d_ws, size_t ws_size,
                              hipStream_t stream)
{
    (void)in_sizes; (void)n_in; (void)out_size; (void)ws_size;

    const float* feats[2] = { (const float*)d_in[0], (const float*)d_in[2] };

    char* ws = (char*)d_ws;
    // workspace layout (bytes)
    float*    D        = (float*)(ws + 0);                      // 64 MB
    float*    p[2];
    p[0]               = (float*)(ws + 67108864);               // 32 MB
    p[1]               = (float*)(ws + 100663296);               // 32 MB
    float*    sq       = (float*)(ws + 134217728);              // 16 KB
    unsigned* tree     = (unsigned*)(ws + 134234112);           // 2 MB
    unsigned* capu     = (unsigned*)(ws + 136331264);           // 4 B (contiguous after tree)
    float*    partials = (float*)(ws + 136331272);              // 4 KB

    const int WASS_BLOCKS = 1024;

    for (int g = 0; g < 2; ++g) {
        // reset per-graph scratch (tree bitmask + cap scalar, contiguous)
        zero_u32_kernel<<<(TREE_U32 + 1 + 255) / 256, 256, 0, stream>>>(tree, TREE_U32 + 1);

        row_sq_kernel<<<(N_PTS + 255) / 256, 256, 0, stream>>>(feats[g], sq);

        // (4096/16)^2 = 65536 tiles, 8 waves/block
        gram_dist_kernel<<<65536 / 8, 256, 0, stream>>>(feats[g], sq, D, capu);

        prim_kernel<<<1, 1024, 0, stream>>>(D, tree);

        persistence_kernel<<<(N_PTS * N_PTS) / 256, 256, 0, stream>>>(D, tree, capu, p[g]);
        pad_zero_kernel<<<(NPAD - NPAIR + 255) / 256, 256, 0, stream>>>(p[g], NPAIR, NPAD);

        // bitonic sort ascending over NPAD elements:
        //   k<=CHUNK fused in LDS; then per-k: global passes j>=CHUNK,
        //   remaining j<CHUNK fused in LDS.
        bitonic_local_sort_kernel<<<NPAD / CHUNK, 256, 0, stream>>>(p[g]);
        for (unsigned k = CHUNK * 2u; k <= NPAD; k <<= 1) {
            for (unsigned j = k >> 1; j >= CHUNK; j >>= 1)
                bitonic_kernel<<<NPAD / 256, 256, 0, stream>>>(p[g], k, j);
            bitonic_local_merge_kernel<<<NPAD / CHUNK, 256, 0, stream>>>(p[g], k);
        }
    }

    wass_partial_kernel<<<WASS_BLOCKS, 256, 0, stream>>>(p[0], p[1], partials);
    wass_final_kernel<<<1, 1024, 0, stream>>>(partials, (float*)d_out, WASS_BLOCKS, 1.0f /*WEIGHT*/);
}